// Lift3DEncoder_75453985456559
// MI455X (gfx1250) — compile-verified
//
#include <hip/hip_runtime.h>
#include <hip/hip_bf16.h>

// ---------------------------------------------------------------------------
// Lift3D encoder: unproject 6 cameras x 64x112 feature pixels into a 64^3
// voxel grid (C=384) with segment-mean pooling.
//
// Shapes (fixed by the reference):
//   image_features (1,6,384,64,112) f32   feature_mask (1,6,64,112) f32
//   depth          (1,6,256,448,1)  f32   camera_pose  (1,6,4,4)    f32
//   camera_intrinsic (1,6,6) f32          neck_voxel_sizes (1,3)    f32
//   out            (1,64,64,64,384) f32 = 100,663,296 floats
// ---------------------------------------------------------------------------

#define NCAM   6
#define CCH    384
#define HF     64
#define WF     112
#define PIX_PER_CAM   (HF * WF)        // 7168
#define NPIX          (NCAM * PIX_PER_CAM)  // 43008
#define NVOX          (64 * 64 * 64)   // 262144
#define OUT_FLOATS    ((size_t)NVOX * CCH)  // 100,663,296
#define TILES_PER_CAM (PIX_PER_CAM / 16)    // 448
#define NTILES        (NCAM * TILES_PER_CAM) // 2688

typedef __attribute__((ext_vector_type(2))) float v2f;
typedef __attribute__((ext_vector_type(8))) float v8f;

#if defined(__has_builtin)
#if __has_builtin(__builtin_amdgcn_wmma_f32_16x16x4_f32)
#define HAVE_WMMA_F32_16X16X4 1
#endif
#endif

// ---------------------------------------------------------------------------
// Kernel 1: zero the output grid (403 MB) and the per-voxel counters.
// Memory-bound floor of the whole problem; use b128 stores.
// ---------------------------------------------------------------------------
__global__ __launch_bounds__(256) void lift_zero(float* __restrict__ out,
                                                 float* __restrict__ cnts) {
  const size_t stride = (size_t)gridDim.x * blockDim.x;
  size_t idx = (size_t)blockIdx.x * blockDim.x + threadIdx.x;
  float4 z4 = make_float4(0.f, 0.f, 0.f, 0.f);
  float4* o4 = reinterpret_cast<float4*>(out);
  const size_t n_out4 = OUT_FLOATS / 4;          // 25,165,824
  for (size_t i = idx; i < n_out4; i += stride) o4[i] = z4;
  float4* c4 = reinterpret_cast<float4*>(cnts);
  const size_t n_cnt4 = NVOX / 4;                // 65,536
  for (size_t i = idx; i < n_cnt4; i += stride) c4[i] = z4;
}

// ---------------------------------------------------------------------------
// Kernel 2: per-pixel voxel rank.  One wave32 handles a tile of 16 pixels of
// one camera.  The 4x4 pose transform of the 16 homogeneous cam-space points
// is done with V_WMMA_F32_16X16X4_F32 (A = 16x4 cam coords, B = pose^T padded
// to 4x16, D[:,0:3] = world xyz) — exact f32 arithmetic, matches FMA.
//
// ISA VGPR layouts (CDNA5 ISA):
//   A 16x4 f32 : lanes 0-15 -> M=lane, {v0,v1}={K0,K1}; lanes 16-31 -> {K2,K3}
//   B 4x16 f32 : lanes 0-15 -> N=lane, {v0,v1}={K0,K1}; lanes 16-31 -> {K2,K3}
//   D 16x16 f32: lane = N (mod 16), VGPR r -> M = r + 8*(lane/16)
// ---------------------------------------------------------------------------
__global__ __launch_bounds__(256) void lift_rank(
    const float* __restrict__ depth,   // (6,256,448)
    const float* __restrict__ pose,    // (6,4,4) row-major
    const float* __restrict__ intr,    // (6,6)
    const float* __restrict__ nvs,     // (3,)
    const float* __restrict__ fmask,   // (6,64,112)
    int* __restrict__ ranks,           // (43008,)
    float* __restrict__ cnts)          // (262144,)
{
  __shared__ float xyz[8][16][4];      // 8 waves x 16 pixels x (x,y,z,pad)

  const int lane = threadIdx.x & 31;
  const int wave = threadIdx.x >> 5;
  const int tile = blockIdx.x * 8 + wave;        // 336 blocks * 8 = 2688
  const int cam  = tile / TILES_PER_CAM;
  const int pixInCam = (tile % TILES_PER_CAM) * 16 + (lane & 15);
  const int h = pixInCam / WF;
  const int w = pixInCam % WF;

  // nearest depth sample: src index = floor(dst*4) since H/Hf = W/Wf = 4
  const float d0 = depth[(size_t)(cam * 256 + h * 4) * 448 + w * 4];
  const bool  sky = (d0 >= 150.0f) || (d0 <= 0.0f);
  const float dr  = sky ? 1.0e6f : d0;

  // intrinsics scaled by 1/4 (sx = sy = 4)
  const float fx = intr[cam * 6 + 0] * 0.25f;
  const float fy = intr[cam * 6 + 1] * 0.25f;
  const float cx = intr[cam * 6 + 2] * 0.25f;
  const float cy = intr[cam * 6 + 3] * 0.25f;
  const float xn = ((float)w - cx) / fx;
  const float yn = ((float)h - cy) / fy;

#if defined(HAVE_WMMA_F32_16X16X4)
  // A: cam-space homogeneous point per pixel M
  v2f a;
  a.x = (lane < 16) ? xn * dr : dr;
  a.y = (lane < 16) ? yn * dr : 1.0f;
  // B: pose^T, columns >= 4 zeroed
  const int Nn = lane & 15;
  const int k0 = (lane < 16) ? 0 : 2;
  v2f b;
  b.x = (Nn < 4) ? pose[cam * 16 + Nn * 4 + k0]     : 0.0f;
  b.y = (Nn < 4) ? pose[cam * 16 + Nn * 4 + k0 + 1] : 0.0f;
  v8f acc = {};
  acc = __builtin_amdgcn_wmma_f32_16x16x4_f32(
      /*neg_a=*/false, a, /*neg_b=*/false, b,
      /*c_mod=*/(short)0, acc, /*reuse_a=*/false, /*reuse_b=*/false);

  // scatter D[:,0:3] (world xyz) to LDS, one slot per pixel
  const int mBase = (lane >> 4) << 3;
  const int Nc = lane & 15;
#pragma unroll
  for (int r = 0; r < 8; ++r) {
    if (Nc < 3) xyz[wave][mBase + r][Nc] = acc[r];
  }
#else
  // scalar fallback: world = pose * cam per pixel
  if (lane < 16) {
    const float cx0 = xn * dr, cy0 = yn * dr, cz0 = dr;
    const float* P = pose + cam * 16;
    xyz[wave][lane][0] = P[0] * cx0 + P[1] * cy0 + P[2]  * cz0 + P[3];
    xyz[wave][lane][1] = P[4] * cx0 + P[5] * cy0 + P[6]  * cz0 + P[7];
    xyz[wave][lane][2] = P[8] * cx0 + P[9] * cy0 + P[10] * cz0 + P[11];
  }
#endif
  __syncthreads();

  if (lane < 16) {
    const float x = xyz[wave][lane][0];
    const float y = xyz[wave][lane][1];
    const float z = xyz[wave][lane][2];
    const float vx = nvs[0], vy = nvs[1], vz = nvs[2];
    // round-half-even matches jnp.round
    const int i = (int)rintf((x - 0.5f * vx) / vx) + 32;
    const int j = (int)rintf((y - 0.5f * vy) / vy) + 32;
    const int k = (int)rintf((z - 0.5f * vz) / vz) + 32;
    const bool in_grid = ((unsigned)i < 64u) & ((unsigned)j < 64u) &
                         ((unsigned)k < 64u);
    const int gpix = cam * PIX_PER_CAM + pixInCam;
    const bool valid = in_grid && (fmask[gpix] > 0.5f);
    const int rank = valid ? (i * 4096 + j * 64 + k) : -1;
    ranks[gpix] = rank;
    if (valid) atomicAdd(&cnts[rank], 1.0f);
  }
}

// ---------------------------------------------------------------------------
// Kernel 3: scatter features.  One block per (cam, h) row: reads along W are
// contiguous 112-float lines per channel (channel stride = 7168 floats),
// atomics are non-returning global_atomic_add_f32 into d_out.
// ---------------------------------------------------------------------------
__global__ __launch_bounds__(128) void lift_scatter(
    const float* __restrict__ feats,   // (6,384,64,112)
    const int* __restrict__ ranks,     // (43008,)
    float* __restrict__ out)           // (262144,384)
{
  const int cam = blockIdx.x / HF;
  const int h   = blockIdx.x % HF;
  const int w   = threadIdx.x;
  if (w >= WF) return;
  const int r = ranks[cam * PIX_PER_CAM + h * WF + w];
  if (r < 0) return;
  const float* __restrict__ src =
      feats + (size_t)cam * CCH * PIX_PER_CAM + (size_t)h * WF + w;
  float* __restrict__ dst = out + (size_t)r * CCH;
#pragma unroll 4
  for (int c = 0; c < CCH; ++c) {
    if (c + 8 < CCH)
      __builtin_prefetch(src + (size_t)(c + 8) * PIX_PER_CAM, 0, 0);
    atomicAdd(dst + c, src[(size_t)c * PIX_PER_CAM]);
  }
}

// ---------------------------------------------------------------------------
// Kernel 4: finalize mean.  count<=1 means divide-by-1 (no-op, zeros stay
// zero), so only voxels with count>=2 are touched (<=43008 of 262144).
// ---------------------------------------------------------------------------
__global__ __launch_bounds__(128) void lift_finalize(
    float* __restrict__ out, const float* __restrict__ cnts) {
  const int v = blockIdx.x;
  const float c = cnts[v];
  if (c < 1.5f) return;
  float* __restrict__ p = out + (size_t)v * CCH;
  for (int i = threadIdx.x; i < CCH; i += 128) p[i] = p[i] / c;
}

// ---------------------------------------------------------------------------
extern "C" void kernel_launch(void* const* d_in, const int* in_sizes, int n_in,
                              void* d_out, int out_size, void* d_ws,
                              size_t ws_size, hipStream_t stream) {
  const float* feats = (const float*)d_in[0];
  const float* fmask = (const float*)d_in[1];
  const float* depth = (const float*)d_in[2];
  const float* pose  = (const float*)d_in[3];
  const float* intr  = (const float*)d_in[4];
  const float* nvs   = (const float*)d_in[5];
  float* out = (float*)d_out;

  // workspace: [0, 172KB) pixel ranks, [256KB, 256KB+1MB) voxel counts
  int*   ranks = (int*)d_ws;
  float* cnts  = (float*)((char*)d_ws + 256 * 1024);

  lift_zero<<<4096, 256, 0, stream>>>(out, cnts);
  lift_rank<<<NTILES / 8, 256, 0, stream>>>(depth, pose, intr, nvs, fmask,
                                            ranks, cnts);
  lift_scatter<<<NCAM * HF, 128, 0, stream>>>(feats, ranks, out);
  lift_finalize<<<NVOX, 128, 0, stream>>>(out, cnts);
}